// TotalLoss_36120674959541
// MI455X (gfx1250) — compile-verified
//
#include <hip/hip_runtime.h>
#include <hip/hip_bf16.h>
#include <float.h>

typedef __attribute__((ext_vector_type(2))) float v2f;
typedef __attribute__((ext_vector_type(8))) float v8f;

#define GAMMA_C 1.0f
#define ALPHA_C 0.5f
#define BETA_C  1.0f
#define THETA_C 1.0f
#define EPS_C   1e-8f

__device__ __forceinline__ float wave_sum32(float v) {
#pragma unroll
  for (int m = 16; m >= 1; m >>= 1) v += __shfl_xor(v, m, 32);
  return v;
}
__device__ __forceinline__ float wave_max32(float v) {
#pragma unroll
  for (int m = 16; m >= 1; m >>= 1) v = fmaxf(v, __shfl_xor(v, m, 32));
  return v;
}

// For each point in P1 (B x n1 x 3): min squared distance to any point of P2 (B x n2 x 3).
// One wave32 owns TWO 16-row strips (32 rows) of P1 and streams P2 in 16-col chunks.
// Core op: V_WMMA_F32_16X16X4_F32 computing the partial form
//   A(16x4) row m = ( sx, sy, sz, 1 )
//   B(4x16) col n = ( -2xx, -2xy, -2xz, |x_n|^2 )
//   C = 0 (inline)
// => D'[m][n] = |x_n|^2 - 2 s.x ;  true dist = max(D' + |s_m|^2, 0), applied after the
// min-reduction (monotonicity lets |s|^2 and the clamp be hoisted out of the loop).
__global__ void __launch_bounds__(256) mindist_wmma_kernel(
    const float* __restrict__ P1, const float* __restrict__ P2,
    int n1, int n2, int B, float* __restrict__ out /* B*n1 */) {
  const int lane = threadIdx.x & 31;
  const int lm   = lane & 15;
  const int half = lane >> 4;
  const int wave = (int)((blockIdx.x * blockDim.x + threadIdx.x) >> 5);
  const int stripsPerBatch = n1 >> 5;  // 32 rows per wave
  if (wave >= B * stripsPerBatch) return;  // wave-uniform exit, EXEC stays all-1 for WMMA
  const int b = wave / stripsPerBatch;
  const int rowbase = (wave % stripsPerBatch) << 5;

  // Load the two A strips (rows rowbase+lm and rowbase+16+lm)
  const float* p0 = P1 + ((size_t)b * n1 + rowbase + lm) * 3;
  const float* p1 = p0 + 48;  // +16 rows * 3 floats
  float s0x = p0[0], s0y = p0[1], s0z = p0[2];
  float s1x = p1[0], s1y = p1[1], s1z = p1[2];
  float sn0 = s0x * s0x + s0y * s0y + s0z * s0z;
  float sn1 = s1x * s1x + s1y * s1y + s1z * s1z;
  v2f a0, a1;  // A layout: VGPR0 = K0|K2, VGPR1 = K1|K3 across lane halves
  a0.x = half ? s0z : s0x;  a0.y = half ? 1.0f : s0y;
  a1.x = half ? s1z : s1x;  a1.y = half ? 1.0f : s1y;

  v8f zero;
#pragma unroll
  for (int i = 0; i < 8; ++i) zero[i] = 0.0f;
  v8f rm0 = zero, rm1 = zero;
#pragma unroll
  for (int i = 0; i < 8; ++i) { rm0[i] = FLT_MAX; rm1[i] = FLT_MAX; }

  const float* colBase = P2 + (size_t)b * n2 * 3;
  for (int cb = 0; cb < n2; cb += 16) {
    const float* q = colBase + (size_t)(cb + lm) * 3;
    float qx = q[0], qy = q[1], qz = q[2];
    float qn = qx * qx + qy * qy + qz * qz;
    v2f bv;  // B layout mirrors A: VGPR0 = K0|K2, VGPR1 = K1|K3 across lane halves
    bv.x = half ? (-2.0f * qz) : (-2.0f * qx);
    bv.y = half ? qn           : (-2.0f * qy);
    v8f d0 = __builtin_amdgcn_wmma_f32_16x16x4_f32(false, a0, false, bv,
                                                   (short)0, zero, false, false);
    v8f d1 = __builtin_amdgcn_wmma_f32_16x16x4_f32(false, a1, false, bv,
                                                   (short)0, zero, false, false);
#pragma unroll
    for (int i = 0; i < 8; ++i) {
      rm0[i] = fminf(rm0[i], d0[i]);
      rm1[i] = fminf(rm1[i], d1[i]);
    }
  }

  // Min over the 16 column-lanes of each half (masks 1,2,4,8 stay within halves).
#pragma unroll
  for (int m = 1; m <= 8; m <<= 1) {
#pragma unroll
    for (int i = 0; i < 8; ++i) {
      rm0[i] = fminf(rm0[i], __shfl_xor(rm0[i], m, 32));
      rm1[i] = fminf(rm1[i], __shfl_xor(rm1[i], m, 32));
    }
  }

  // D tile rows: VGPR i -> row (half*8 + i) of the strip. Fetch that row's |s|^2
  // from lane (half*8 + i) of the own 16-lane segment, add, clamp, store.
  if (lm == 0) {  // lanes 0 and 16 write
    float* o = out + (size_t)b * n1 + rowbase + half * 8;
#pragma unroll
    for (int i = 0; i < 8; ++i) {
      float n0 = __shfl(sn0, half * 8 + i, 16);
      float n1 = __shfl(sn1, half * 8 + i, 16);
      o[i]      = fmaxf(rm0[i] + n0, 0.0f);
      o[i + 16] = fmaxf(rm1[i] + n1, 0.0f);
    }
  }
}

__global__ void init_acc_kernel(float* acc) {
  if (threadIdx.x < 16) acc[threadIdx.x] = 0.0f;
}

// Per (b,m): sum dist_smp, per-batch max, trans/rot/cls terms.
// acc: [0]=sum_smp [1]=sum_org [2]=sum_trans [3]=sum_rot [4]=sum_cls [8..11]=per-batch max
__global__ void reduce_smp_kernel(
    const float* __restrict__ dist_smp, const float* __restrict__ grasp_pred,
    const float* __restrict__ grasp_gt, const float* __restrict__ class_pred,
    const float* __restrict__ class_gt, int M, int total, float* __restrict__ acc) {
  int i = blockIdx.x * blockDim.x + threadIdx.x;
  float dv = 0.0f, trans = 0.0f, rot = 0.0f, cls = 0.0f;
  int b = 0;
  if (i < total) {
    dv = dist_smp[i];
    b = i / M;  // M % 32 == 0 -> wave-uniform

    const float* gp = grasp_pred + (size_t)i * 7;
    float cpx = gp[0], cpy = gp[1], cpz = gp[2];
    float qw = gp[3], qx = gp[4], qy = gp[5], qz = gp[6];
    float inv = 1.0f / sqrtf(fmaxf(qw * qw + qx * qx + qy * qy + qz * qz, EPS_C));
    qw *= inv; qx *= inv; qy *= inv; qz *= inv;

    const float* T = grasp_gt + (size_t)i * 16;  // 4x4 row-major
    float r00 = T[0], r01 = T[1], r02 = T[2],  cgx = T[3];
    float r10 = T[4], r11 = T[5], r12 = T[6],  cgy = T[7];
    float r20 = T[8], r21 = T[9], r22 = T[10], cgz = T[11];
    float gw = 0.5f * sqrtf(fmaxf(1.0f + r00 + r11 + r22, EPS_C));
    float gx = copysignf(0.5f * sqrtf(fmaxf(1.0f + r00 - r11 - r22, EPS_C)), r21 - r12);
    float gy = copysignf(0.5f * sqrtf(fmaxf(1.0f - r00 + r11 - r22, EPS_C)), r02 - r20);
    float gz = copysignf(0.5f * sqrtf(fmaxf(1.0f - r00 - r11 + r22, EPS_C)), r10 - r01);

    float dx = cpx - cgx + 1e-6f, dy = cpy - cgy + 1e-6f, dz = cpz - cgz + 1e-6f;
    trans = sqrtf(dx * dx + dy * dy + dz * dz);
    rot   = 1.0f - (qw * gw + qx * gx + qy * gy + qz * gz);

    float p = fminf(fmaxf(class_pred[i], 1e-7f), 1.0f - 1e-7f);
    float y = class_gt[i];
    cls = -(y * logf(p) + (1.0f - y) * log1pf(-p));
  }
  float s0 = wave_sum32(dv);
  float s1 = wave_sum32(trans);
  float s2 = wave_sum32(rot);
  float s3 = wave_sum32(cls);
  float mx = wave_max32(dv);  // inactive lanes contribute 0; dists are >= 0
  if ((threadIdx.x & 31) == 0) {
    atomicAdd(&acc[0], s0);
    atomicAdd(&acc[2], s1);
    atomicAdd(&acc[3], s2);
    atomicAdd(&acc[4], s3);
    atomicMax((unsigned int*)&acc[8 + b], __float_as_uint(mx));  // bit-order==float-order, x>=0
  }
}

__global__ void reduce_org_kernel(const float* __restrict__ dist_org, int total,
                                  float* __restrict__ acc) {
  int i = blockIdx.x * blockDim.x + threadIdx.x;
  float v = (i < total) ? dist_org[i] : 0.0f;
  float s = wave_sum32(v);
  if ((threadIdx.x & 31) == 0) atomicAdd(&acc[1], s);
}

__global__ void finalize_kernel(const float* __restrict__ acc, const float* __restrict__ temp,
                                int B, int M, int N, float* __restrict__ out) {
  if (threadIdx.x == 0 && blockIdx.x == 0) {
    float invBM = 1.0f / (float)(B * M);
    float invBN = 1.0f / (float)(B * N);
    float maxsum = 0.0f;
    for (int b = 0; b < B; ++b) maxsum += acc[8 + b];
    float sample_loss = acc[0] * invBM + maxsum / (float)B + GAMMA_C * acc[1] * invBN;
    float t = temp[0];
    float reg_loss = acc[2] * invBM + ALPHA_C * (acc[3] * invBM);
    out[0] = THETA_C * sample_loss + t * t + BETA_C * reg_loss + acc[4] * invBM;
  }
}

extern "C" void kernel_launch(void* const* d_in, const int* in_sizes, int n_in,
                              void* d_out, int out_size, void* d_ws, size_t ws_size,
                              hipStream_t stream) {
  const float* xyz        = (const float*)d_in[0];  // (B,N,3)
  const float* sample_xyz = (const float*)d_in[1];  // (B,M,3)
  const float* temp       = (const float*)d_in[2];  // scalar
  const float* grasp_pred = (const float*)d_in[3];  // (B,M,7)
  const float* grasp_gt   = (const float*)d_in[4];  // (B,M,4,4)
  const float* class_pred = (const float*)d_in[5];  // (B,M)
  const float* class_gt   = (const float*)d_in[6];  // (B,M)

  const int B = 4;
  const int N = in_sizes[0] / (B * 3);
  const int M = in_sizes[1] / (B * 3);

  float* acc      = (float*)d_ws;                 // 16 floats
  float* dist_smp = acc + 16;                     // B*M
  float* dist_org = dist_smp + (size_t)B * M;     // B*N

  init_acc_kernel<<<1, 32, 0, stream>>>(acc);

  int waves1 = (B * M) / 32;   // sample -> xyz, 32 rows/wave
  mindist_wmma_kernel<<<(waves1 * 32 + 255) / 256, 256, 0, stream>>>(
      sample_xyz, xyz, M, N, B, dist_smp);
  int waves2 = (B * N) / 32;   // xyz -> sample
  mindist_wmma_kernel<<<(waves2 * 32 + 255) / 256, 256, 0, stream>>>(
      xyz, sample_xyz, N, M, B, dist_org);

  int totBM = B * M;
  reduce_smp_kernel<<<(totBM + 255) / 256, 256, 0, stream>>>(
      dist_smp, grasp_pred, grasp_gt, class_pred, class_gt, M, totBM, acc);
  int totBN = B * N;
  reduce_org_kernel<<<(totBN + 255) / 256, 256, 0, stream>>>(dist_org, totBN, acc);

  finalize_kernel<<<1, 32, 0, stream>>>(acc, temp, B, M, N, (float*)d_out);
}